// CodeBook_22411139350887
// MI455X (gfx1250) — compile-verified
//
#include <hip/hip_runtime.h>

typedef float v2f __attribute__((ext_vector_type(2)));
typedef float v4f __attribute__((ext_vector_type(4)));
typedef float v8f __attribute__((ext_vector_type(8)));

#define D_DIM   256
#define K_CODES 1024
#define N_ROWS  32768                 // 32 * 32 * 32
#define Z_ELEMS (N_ROWS * D_DIM)      // 8388608
#define HW      1024                  // 32*32 positions per image
#define NTILES  (N_ROWS / 16)         // 2048 row tiles
#define CTILES  (K_CODES / 16)        // 64 code tiles
#define DCHUNKS (D_DIM / 4)           // 64 k-chunks of 4 for f32 WMMA
#define PCHUNKS (DCHUNKS / 2)         // 32 paired chunks (one b128 each)

// ---------------- prep: pack B fragment PAIRS + codebook norms ----------------
// Bpack float index: ct*4096 + pc*128 + lane*4 + q   (q in 0..3)
//   dc = pc*2 + (q>>1), j = q&1
//   holds W[code = ct*16 + (lane&15)][d = dc*4 + 2*(lane>>4) + j]
// so one b128 per lane supplies the B fragments for chunks (2*pc, 2*pc+1),
// matching the fp32 WMMA B lane layout (N = lane&15, half selects K pair).
__global__ __launch_bounds__(256) void vq_prep(const float* __restrict__ W,
                                               float* __restrict__ Bpack,
                                               float* __restrict__ Wsq) {
    if (blockIdx.x < 1024) {
        int t    = blockIdx.x * 256 + threadIdx.x;      // 0..262143
        int q    = t & 3;
        int lane = (t >> 2) & 31;
        int pc   = (t >> 7) & 31;
        int ct   = t >> 12;
        int dc   = pc * 2 + (q >> 1);
        int j    = q & 1;
        int code = ct * 16 + (lane & 15);
        int d    = dc * 4 + 2 * (lane >> 4) + j;
        Bpack[t] = W[code * D_DIM + d];
    } else {
        int k = (blockIdx.x - 1024) * 256 + threadIdx.x; // 0..1023
        float s = 0.f;
        for (int i = 0; i < D_DIM; ++i) {
            float w = W[k * D_DIM + i];
            s = __builtin_fmaf(w, w, s);
        }
        Wsq[k] = s;
    }
}

// ---------------- main: one wave per 16-row tile, WMMA fp32 ----------------
__global__ __launch_bounds__(32) void vq_main(const float* __restrict__ X,
                                              const float* __restrict__ W,
                                              const float* __restrict__ Bpack,
                                              const float* __restrict__ Wsq,
                                              float* __restrict__ outZ,
                                              float* __restrict__ outIdx) {
    const int rowTile = blockIdx.x;            // 0..2047
    const int lane    = threadIdx.x;           // 0..31
    const int l       = lane & 15;
    const int half    = lane >> 4;

    const int b   = rowTile >> 6;              // image index (1024 rows / image)
    const int hw0 = (rowTile & 63) * 16;       // base spatial position

    // A tile: 16 rows x 256 d, held in registers as 64 fp32 fragments.
    // A[M = lane&15][K = 2*half + j]; memory: X[b][d][hw0 + M], contiguous in M.
    const float* pa = X + (size_t)b * (D_DIM * HW) + (2 * half) * HW + hw0 + l;
    v2f a[DCHUNKS];
#pragma unroll
    for (int dc = 0; dc < DCHUNKS; ++dc) {
        a[dc].x = pa[dc * 4 * HW];
        a[dc].y = pa[dc * 4 * HW + HW];
    }

    float minv[8];
    int   mini[8];
#pragma unroll
    for (int j = 0; j < 8; ++j) { minv[j] = 3.4e38f; mini[j] = 0; }

    for (int ct = 0; ct < CTILES; ++ct) {
        // one b128 per lane per pair -> feeds two WMMAs on independent accumulators
        const v4f* pb = (const v4f*)Bpack + ct * (PCHUNKS * 32) + lane;
        v8f acc0 = {0.f, 0.f, 0.f, 0.f, 0.f, 0.f, 0.f, 0.f};
        v8f acc1 = {0.f, 0.f, 0.f, 0.f, 0.f, 0.f, 0.f, 0.f};
#pragma unroll
        for (int pc = 0; pc < PCHUNKS; ++pc) {
            v4f bb = pb[pc * 32];
            v2f b0 = {bb.x, bb.y};
            v2f b1 = {bb.z, bb.w};
            acc0 = __builtin_amdgcn_wmma_f32_16x16x4_f32(
                false, a[2 * pc], false, b0, (short)0, acc0, false, false);
            acc1 = __builtin_amdgcn_wmma_f32_16x16x4_f32(
                false, a[2 * pc + 1], false, b1, (short)0, acc1, false, false);
        }
        const int   code = ct * 16 + l;
        const float wq   = Wsq[code];
#pragma unroll
        for (int j = 0; j < 8; ++j) {
            float dot = acc0[j] + acc1[j];
            float s = __builtin_fmaf(-2.f, dot, wq);     // ||w||^2 - 2 z.w
            if (s < minv[j]) { minv[j] = s; mini[j] = code; }
        }
    }

    // Cross-lane argmin over the 16 codes living across lanes of each half.
#pragma unroll
    for (int j = 0; j < 8; ++j) {
        float v = minv[j];
        int   i = mini[j];
#pragma unroll
        for (int off = 1; off < 16; off <<= 1) {
            float ov = __shfl_xor(v, off, 32);
            int   oi = __shfl_xor(i, off, 32);
            if (ov < v || (ov == v && oi < i)) { v = ov; i = oi; }
        }
        minv[j] = v;
        mini[j] = i;
    }

    // idx output (as float): row r = j + 8*half
    if (l == 0) {
#pragma unroll
        for (int j = 0; j < 8; ++j)
            outIdx[rowTile * 16 + 8 * half + j] = (float)mini[j];
    }

    // z_q output: reference does W[idx].reshape(x.shape) as a VIEW, i.e. a flat
    // row-major copy of the selected codebook rows.
#pragma unroll
    for (int r = 0; r < 16; ++r) {
        int code = __shfl(mini[r & 7], (r >> 3) * 16, 32);
        const float4* src = (const float4*)(W + (size_t)code * D_DIM);
        float4* dst = (float4*)(outZ + ((size_t)rowTile * 16 + r) * D_DIM);
        dst[lane]      = src[lane];
        dst[lane + 32] = src[lane + 32];
    }
}

// ---------------- loss: deterministic two-pass reduction ----------------
__global__ __launch_bounds__(256) void loss_partial(const float* __restrict__ zq,
                                                    const float* __restrict__ x,
                                                    float* __restrict__ partial) {
    const int base = blockIdx.x * 8192 + threadIdx.x;
    float s = 0.f;
#pragma unroll
    for (int i = 0; i < 32; ++i) {
        float d = zq[base + i * 256] - x[base + i * 256];
        s = __builtin_fmaf(d, d, s);
    }
#pragma unroll
    for (int off = 16; off > 0; off >>= 1) s += __shfl_xor(s, off, 32);
    __shared__ float wsum[8];
    if ((threadIdx.x & 31) == 0) wsum[threadIdx.x >> 5] = s;
    __syncthreads();
    if (threadIdx.x == 0) {
        float t = 0.f;
        for (int w = 0; w < 8; ++w) t += wsum[w];
        partial[blockIdx.x] = t;
    }
}

__global__ __launch_bounds__(32) void loss_final(const float* __restrict__ partial,
                                                 float* __restrict__ out) {
    float s = 0.f;
    for (int i = 0; i < 32; ++i) s += partial[threadIdx.x * 32 + i];
#pragma unroll
    for (int off = 16; off > 0; off >>= 1) s += __shfl_xor(s, off, 32);
    if (threadIdx.x == 0)
        out[0] = s * ((1.f + 0.25f) / (float)Z_ELEMS);   // mean + BETA*mean collapse
}

// ---------------- host launcher ----------------
extern "C" void kernel_launch(void* const* d_in, const int* in_sizes, int n_in,
                              void* d_out, int out_size, void* d_ws, size_t ws_size,
                              hipStream_t stream) {
    (void)in_sizes; (void)n_in; (void)out_size; (void)ws_size;
    const float* X = (const float*)d_in[0];   // [32,256,32,32]
    const float* W = (const float*)d_in[1];   // [1024,256]

    float* out   = (float*)d_out;
    float* outZ  = out;                 // 8388608 floats
    float* outI  = out + Z_ELEMS;       // 32768 floats
    float* outL  = outI + N_ROWS;       // 1 float

    float* Bpack   = (float*)d_ws;                  // 262144 floats (1 MB)
    float* Wsq     = Bpack + (K_CODES * D_DIM);     // 1024 floats
    float* partial = Wsq + K_CODES;                 // 1024 floats

    vq_prep<<<1028, 256, 0, stream>>>(W, Bpack, Wsq);
    vq_main<<<NTILES, 32, 0, stream>>>(X, W, Bpack, Wsq, outZ, outI);
    loss_partial<<<1024, 256, 0, stream>>>(outZ, X, partial);
    loss_final<<<1, 32, 0, stream>>>(partial, outL);
}